// Wordwise_16922171146747
// MI455X (gfx1250) — compile-verified
//
#include <hip/hip_runtime.h>

typedef __attribute__((ext_vector_type(2))) float v2f;
typedef __attribute__((ext_vector_type(8))) float v8f;

#define B_  32
#define C_  80
#define T_  2048
#define W_  128
#define H_  512
#define K_  5
#define CK_  (C_ * K_)    // 400
#define CSTR 84           // LDS channel-stride (floats). 80 used, padded to 84:
                          // bank stride 20 (mod 64) -> lanes 0-15 hit banks
                          // =0 mod 4, lanes 16-31 (c+2) hit banks =2 mod 4:
                          // conflict-free ds_load_b64 B-fragment reads.

// ---------------------------------------------------------------------------
// Kernel 1: encoder conv as WMMA GEMM + bias + ReLU.
//   E[bl, h, t] = relu( sum_{c,k} feat[b0+bl, c, t+k-2] * encw[h, c, k] + encb[h] )
//
// Block = 256 threads = 8 waves sharing one 64-wide t-slice. The zero-padded
// feature tile (68 time cols x 80 ch) is staged in LDS *transposed*
// (tileB[dt*CSTR + c]) so each WMMA B fragment {c, c+1} is one aligned
// ds_load_b64 into an even VGPR pair — no register shuffles. Each wave owns a
// different 16-row H-tile -> 8x B reuse from LDS. SAME padding is resolved at
// staging time, so the MMA loop is compare-free and EXEC is all-1s at every
// V_WMMA_F32_16X16X4_F32 (grid is exact, no early returns).
// ---------------------------------------------------------------------------
__global__ __launch_bounds__(256) void enc_gemm_wmma(
    const float* __restrict__ feat, const float* __restrict__ encw,
    const float* __restrict__ encb, float* __restrict__ E, int b0)
{
    __shared__ float tileB[68 * CSTR];   // 68*84*4 = 22.3 KiB

    const int blk = blockIdx.x;
    const int tT  = blk % (T_ / 64);                 // 32 t-tiles
    const int hG  = (blk / (T_ / 64)) % (H_ / 128);  // 4 groups of 8 H-tiles
    const int bl  = blk / ((T_ / 64) * (H_ / 128));
    const int b   = b0 + bl;

    const int wave = threadIdx.x >> 5;
    const int lane = threadIdx.x & 31;
    const int t0   = tT * 64;
    const int m0   = (hG * 8 + wave) * 16;     // H-tile base row for this wave

    const float* featB = feat + (size_t)b * C_ * T_;

    // ---- stage zero-padded tile, transposed: col dt=0..67 <-> ts=t0-2..t0+65
    for (int i = threadIdx.x; i < C_ * 68; i += 256) {
        const int c  = i / 68;
        const int dt = i - c * 68;           // consecutive threads: coalesced ts
        const int ts = t0 - 2 + dt;
        float v = 0.0f;
        if (ts >= 0 && ts < T_) v = featB[c * T_ + ts];
        tileB[dt * CSTR + c] = v;
    }
    __syncthreads();

    // ---- WMMA fragment coordinates (wave32, f32 16x16x4 layouts)
    const int mrow = lane & 15;        // A row
    const int hi   = lane >> 4;        // half-wave
    const int koff = hi * 2;           // K = vgpr + 2*hi for both A and B
    const int n    = lane & 15;        // B/C/D column

    const float* Wrow = encw + (size_t)(m0 + mrow) * CK_;  // encw[h][c][k]

    v8f acc[4] = { v8f{}, v8f{}, v8f{}, v8f{} };

    #pragma unroll
    for (int kk = 0; kk < K_; ++kk) {              // tap
        #pragma unroll 4
        for (int cb = 0; cb < C_; cb += 4) {       // 4 channels per WMMA K-step
            // A fragment: reduction index ck' = kk*80 + cb + koff (+1)
            v2f a;
            a.x = Wrow[(cb + koff    ) * K_ + kk];
            a.y = Wrow[(cb + koff + 1) * K_ + kk];

            // B fragment: channels {cb+koff, cb+koff+1} at time col j*16+n+kk
            // -> adjacent floats in LDS, one aligned ds_load_b64 each.
            const int col = cb + koff;
            #pragma unroll
            for (int j = 0; j < 4; ++j) {
                const v2f bf =
                    *(const v2f*)&tileB[(j * 16 + n + kk) * CSTR + col];
                acc[j] = __builtin_amdgcn_wmma_f32_16x16x4_f32(
                             false, a, false, bf, (short)0, acc[j],
                             false, false);
            }
        }
    }

    // ---- bias + ReLU + store (C/D layout: M = i + 8*hi, N = lane&15)
    #pragma unroll
    for (int j = 0; j < 4; ++j) {
        #pragma unroll
        for (int i = 0; i < 8; ++i) {
            const int h = m0 + i + 8 * hi;
            float v = acc[j][i] + encb[h];
            v = v > 0.0f ? v : 0.0f;
            E[((size_t)bl * H_ + h) * T_ + (t0 + j * 16 + n)] = v;
        }
    }
}

// ---------------------------------------------------------------------------
// Kernel 2: masked mean pooling per word (generic bounds; matches reference
// mask semantics: t in [start,end) intersected with [0,T), count=max(cnt,1),
// invalid words (w >= word_lengths[b]) contribute zero).
// ---------------------------------------------------------------------------
__global__ void pool_words(const float* __restrict__ E,
                           const int* __restrict__ bounds,
                           const int* __restrict__ wlen,
                           float* __restrict__ WE, int b0, int nb)
{
    const int idx = blockIdx.x * blockDim.x + threadIdx.x;
    if (idx >= nb * H_ * W_) return;
    const int w  = idx % W_;
    const int h  = (idx / W_) % H_;
    const int bl = idx / (W_ * H_);
    const int b  = b0 + bl;

    const int start = bounds[((size_t)b * 2 + 0) * W_ + w];
    const int end   = bounds[((size_t)b * 2 + 1) * W_ + w];
    const int s = start < 0 ? 0 : start;
    const int e = end > T_ ? T_ : end;

    float sum = 0.0f;
    int   cnt = 0;
    if (w < wlen[b]) {
        const float* row = E + ((size_t)bl * H_ + h) * T_;
        for (int t = s; t < e; ++t) { sum += row[t]; ++cnt; }
    }
    const float c = (cnt < 1) ? 1.0f : (float)cnt;
    WE[((size_t)b * H_ + h) * W_ + w] = sum / c;
}

// ---------------------------------------------------------------------------
// Kernel 3: decoder conv1d over the word axis: out[b, w] = dec_b +
//   sum_h sum_k WE[b, h, w+k-2] * dec_w[h, k]   (SAME padding)
// 4096 outputs x 2560 MACs — negligible; one thread per output.
// ---------------------------------------------------------------------------
__global__ void dec_conv(const float* __restrict__ WE,
                         const float* __restrict__ decw,
                         const float* __restrict__ decb,
                         float* __restrict__ out)
{
    const int idx = blockIdx.x * blockDim.x + threadIdx.x;
    if (idx >= B_ * W_) return;
    const int w = idx % W_;
    const int b = idx / W_;

    float sum = decb[0];
    const float* we = WE + (size_t)b * H_ * W_;
    for (int h = 0; h < H_; ++h) {
        const float* row = we + (size_t)h * W_;
        #pragma unroll
        for (int k = 0; k < K_; ++k) {
            const int ws = w + k - 2;
            const float val = (ws >= 0 && ws < W_) ? row[ws] : 0.0f;
            sum += val * decw[h * K_ + k];
        }
    }
    out[idx] = sum;
}

// ---------------------------------------------------------------------------
// Launch. Workspace layout: [ WE : B*H*W floats | E : nb*H*T floats ].
// nb (batches per chunk) is derived from ws_size — deterministic across calls.
// ---------------------------------------------------------------------------
extern "C" void kernel_launch(void* const* d_in, const int* in_sizes, int n_in,
                              void* d_out, int out_size, void* d_ws, size_t ws_size,
                              hipStream_t stream)
{
    (void)in_sizes; (void)n_in; (void)out_size;
    const float* feat   = (const float*)d_in[0];
    const int*   bounds = (const int*)  d_in[1];
    const int*   wlen   = (const int*)  d_in[2];
    const float* encw   = (const float*)d_in[3];
    const float* encb   = (const float*)d_in[4];
    const float* decw   = (const float*)d_in[5];
    const float* decb   = (const float*)d_in[6];
    float* out = (float*)d_out;

    const size_t weBytes    = (size_t)B_ * H_ * W_ * sizeof(float);  // 8 MiB
    const size_t ebPerBatch = (size_t)H_ * T_ * sizeof(float);       // 4 MiB / batch

    float* WE = (float*)d_ws;
    float* E  = (float*)((char*)d_ws + weBytes);

    size_t avail = ws_size > weBytes ? (ws_size - weBytes) : 0;
    int nbMax = (int)(avail / ebPerBatch);
    if (nbMax < 1)  nbMax = 1;     // last-resort: assume at least 12 MiB of ws
    if (nbMax > B_) nbMax = B_;

    for (int b0 = 0; b0 < B_; b0 += nbMax) {
        const int nb = (B_ - b0) < nbMax ? (B_ - b0) : nbMax;
        // blocks: nb * (T/64 t-tiles) * (H/128 groups of 8 H-tiles)
        enc_gemm_wmma<<<dim3(nb * (T_ / 64) * (H_ / 128)), dim3(256), 0, stream>>>(
            feat, encw, encb, E, b0);
        const int nPool = nb * H_ * W_;
        pool_words<<<dim3((nPool + 255) / 256), dim3(256), 0, stream>>>(
            E, bounds, wlen, WE, b0, nb);
    }
    dec_conv<<<dim3((B_ * W_ + 255) / 256), dim3(256), 0, stream>>>(
        WE, decw, decb, out);
}